// TemporalAttn_8418135900508
// MI455X (gfx1250) — compile-verified
//
#include <hip/hip_runtime.h>
#include <math.h>

typedef __attribute__((ext_vector_type(2))) float v2f;
typedef __attribute__((ext_vector_type(8))) float v8f;

#define BATCH 32
#define TLEN  8192
#define HDIM  256
#define CHUNK 128
#define NCHUNK (TLEN / CHUNK)   // 64

// ---------------------------------------------------------------------------
// k1: v[b,h] = sum_o h_t[b,o] * W1[o,h]   (GEMM M=32, N=256, K=256, fp32 WMMA)
// One wave per 16x16 D tile. blockIdx.x -> (mtile, ntile).
// ---------------------------------------------------------------------------
__global__ __launch_bounds__(32) void k1_compute_v(const float* __restrict__ hs,
                                                   const float* __restrict__ W1,
                                                   float* __restrict__ v_ws) {
    const int lane  = threadIdx.x;
    const int mtile = blockIdx.x >> 4;    // 0..1
    const int ntile = blockIdx.x & 15;    // 0..15
    const int mrow  = mtile * 16 + (lane & 15);       // batch row 0..31
    const int half  = lane >> 4;                      // 0: K+0/1, 1: K+2/3
    const int n     = ntile * 16 + (lane & 15);

    const float* ht = hs + ((size_t)mrow * TLEN + (TLEN - 1)) * HDIM; // h_t[mrow]

    v8f c = {};
    for (int kb = 0; kb < HDIM; kb += 4) {
        const int k0 = kb + (half ? 2 : 0);
        const int k1 = k0 + 1;
        v2f a, b;
        a.x = ht[k0];           a.y = ht[k1];           // A 16x4 fragment
        b.x = W1[k0 * HDIM + n]; b.y = W1[k1 * HDIM + n]; // B 4x16 fragment
        c = __builtin_amdgcn_wmma_f32_16x16x4_f32(false, a, false, b,
                                                  (short)0, c, false, false);
    }
    const int mbase = mtile * 16 + (half ? 8 : 0);
#pragma unroll
    for (int i = 0; i < 8; ++i)
        v_ws[(mbase + i) * HDIM + n] = c[i];
}

// ---------------------------------------------------------------------------
// k2: score[b,t] = dot(hs[b,t,:], v[b,:])  -- streaming pass 1 over 256 MB.
// 8 waves/block, one wave handles 8 t values, coalesced float4-pair loads.
// ---------------------------------------------------------------------------
__global__ __launch_bounds__(256) void k2_score(const float* __restrict__ hs,
                                                const float* __restrict__ v_ws,
                                                float* __restrict__ score) {
    __shared__ float sv[HDIM];
    const int b   = blockIdx.y;
    const int tid = threadIdx.x;
    sv[tid] = v_ws[b * HDIM + tid];
    __syncthreads();

    const int wave  = tid >> 5;
    const int lane  = tid & 31;
    const int tbase = blockIdx.x * 64 + wave * 8;
    const float* base = hs + (size_t)b * TLEN * HDIM;

    float vv[8];
#pragma unroll
    for (int j = 0; j < 8; ++j) vv[j] = sv[lane * 8 + j];

    for (int i = 0; i < 8; ++i) {
        const int t = tbase + i;
        const float4* p = (const float4*)(base + (size_t)t * HDIM + lane * 8);
        const float4 x0 = p[0];
        const float4 x1 = p[1];
        float s = x0.x * vv[0] + x0.y * vv[1] + x0.z * vv[2] + x0.w * vv[3]
                + x1.x * vv[4] + x1.y * vv[5] + x1.z * vv[6] + x1.w * vv[7];
#pragma unroll
        for (int off = 16; off > 0; off >>= 1)
            s += __shfl_xor(s, off, 32);
        if (lane == 0) score[b * TLEN + t] = s;
    }
}

// ---------------------------------------------------------------------------
// k3: attention_weights[b,:] = softmax(score[b,:]) over TLEN
// ---------------------------------------------------------------------------
__global__ __launch_bounds__(1024) void k3_softmax(const float* __restrict__ score,
                                                   float* __restrict__ aw) {
    __shared__ float red[1024];
    const int b   = blockIdx.x;
    const int tid = threadIdx.x;
    const float* s = score + b * TLEN;

    float loc[8];
    float m = -INFINITY;
#pragma unroll
    for (int j = 0; j < 8; ++j) {
        loc[j] = s[tid + j * 1024];
        m = fmaxf(m, loc[j]);
    }
    red[tid] = m; __syncthreads();
    for (int off = 512; off > 0; off >>= 1) {
        if (tid < off) red[tid] = fmaxf(red[tid], red[tid + off]);
        __syncthreads();
    }
    m = red[0]; __syncthreads();

    float sum = 0.f;
#pragma unroll
    for (int j = 0; j < 8; ++j) {
        loc[j] = expf(loc[j] - m);
        sum += loc[j];
    }
    red[tid] = sum; __syncthreads();
    for (int off = 512; off > 0; off >>= 1) {
        if (tid < off) red[tid] += red[tid + off];
        __syncthreads();
    }
    const float inv = 1.0f / red[0];
#pragma unroll
    for (int j = 0; j < 8; ++j)
        aw[b * TLEN + tid + j * 1024] = loc[j] * inv;
}

// ---------------------------------------------------------------------------
// k4: partial[b,chunk,h] = sum_{t in chunk} aw[b,t] * hs[b,t,h]
// streaming pass 2; deterministic (no float atomics).
// ---------------------------------------------------------------------------
__global__ __launch_bounds__(256) void k4_context_partial(const float* __restrict__ hs,
                                                          const float* __restrict__ aw,
                                                          float* __restrict__ partial) {
    __shared__ float sw[CHUNK];
    const int b     = blockIdx.y;
    const int chunk = blockIdx.x;
    const int tid   = threadIdx.x;
    const int tbase = chunk * CHUNK;
    if (tid < CHUNK) sw[tid] = aw[b * TLEN + tbase + tid];
    __syncthreads();

    const float* base = hs + ((size_t)b * TLEN + tbase) * HDIM + tid;
    float acc = 0.f;
#pragma unroll 4
    for (int i = 0; i < CHUNK; ++i)
        acc += sw[i] * base[(size_t)i * HDIM];

    partial[((size_t)b * NCHUNK + chunk) * HDIM + tid] = acc;
}

// k4b: ctx[b,h] = sum_chunk partial[b,chunk,h]
__global__ __launch_bounds__(256) void k4b_reduce(const float* __restrict__ partial,
                                                  float* __restrict__ ctx) {
    const int b = blockIdx.x;
    const int h = threadIdx.x;
    float acc = 0.f;
    for (int c = 0; c < NCHUNK; ++c)
        acc += partial[((size_t)b * NCHUNK + c) * HDIM + h];
    ctx[b * HDIM + h] = acc;
}

// ---------------------------------------------------------------------------
// k5: av[b,n] = tanh( sum_k concat(ctx,h_t)[b,k] * W2[n,k] )
// GEMM M=32, N=256, K=512 via fp32 WMMA; K split at 256 (kb%4==0 so uniform).
// ---------------------------------------------------------------------------
__global__ __launch_bounds__(32) void k5_final(const float* __restrict__ hs,
                                               const float* __restrict__ ctx,
                                               const float* __restrict__ W2,
                                               float* __restrict__ av) {
    const int lane  = threadIdx.x;
    const int mtile = blockIdx.x >> 4;
    const int ntile = blockIdx.x & 15;
    const int mrow  = mtile * 16 + (lane & 15);
    const int half  = lane >> 4;
    const int n     = ntile * 16 + (lane & 15);

    const float* ht = hs + ((size_t)mrow * TLEN + (TLEN - 1)) * HDIM;
    const float* ca = ctx + mrow * HDIM;
    const float* w2row = W2 + (size_t)n * (2 * HDIM);

    v8f c = {};
    for (int kb = 0; kb < HDIM; kb += 4) {           // k in [0,256): context half
        const int k0 = kb + (half ? 2 : 0), k1 = k0 + 1;
        v2f a, b;
        a.x = ca[k0];     a.y = ca[k1];
        b.x = w2row[k0];  b.y = w2row[k1];
        c = __builtin_amdgcn_wmma_f32_16x16x4_f32(false, a, false, b,
                                                  (short)0, c, false, false);
    }
    for (int kb = 0; kb < HDIM; kb += 4) {           // k in [256,512): h_t half
        const int k0 = kb + (half ? 2 : 0), k1 = k0 + 1;
        v2f a, b;
        a.x = ht[k0];           a.y = ht[k1];
        b.x = w2row[HDIM + k0]; b.y = w2row[HDIM + k1];
        c = __builtin_amdgcn_wmma_f32_16x16x4_f32(false, a, false, b,
                                                  (short)0, c, false, false);
    }
    const int mbase = mtile * 16 + (half ? 8 : 0);
#pragma unroll
    for (int i = 0; i < 8; ++i)
        av[(mbase + i) * HDIM + n] = tanhf(c[i]);
}

// ---------------------------------------------------------------------------
extern "C" void kernel_launch(void* const* d_in, const int* in_sizes, int n_in,
                              void* d_out, int out_size, void* d_ws, size_t ws_size,
                              hipStream_t stream) {
    const float* hs = (const float*)d_in[0];   // (32, 8192, 256)
    const float* W1 = (const float*)d_in[1];   // (256, 256)
    const float* W2 = (const float*)d_in[2];   // (256, 512)

    float* out = (float*)d_out;
    float* av  = out;                 // attention_vector: 32*256
    float* aw  = out + BATCH * HDIM;  // attention_weights: 32*8192

    char* ws = (char*)d_ws;
    float* v_ws    = (float*)ws;                                   // 32 KB
    float* score   = (float*)(ws + 32 * 1024);                     // 1 MB
    float* partial = (float*)(ws + 32 * 1024
                              + (size_t)BATCH * TLEN * 4);         // 2 MB
    float* ctx     = (float*)(ws + 32 * 1024
                              + (size_t)BATCH * TLEN * 4
                              + (size_t)BATCH * NCHUNK * HDIM * 4);// 32 KB

    k1_compute_v       <<<32, 32, 0, stream>>>(hs, W1, v_ws);
    k2_score           <<<dim3(TLEN / 64, BATCH), 256, 0, stream>>>(hs, v_ws, score);
    k3_softmax         <<<BATCH, 1024, 0, stream>>>(score, aw);
    k4_context_partial <<<dim3(NCHUNK, BATCH), 256, 0, stream>>>(hs, aw, partial);
    k4b_reduce         <<<BATCH, 256, 0, stream>>>(partial, ctx);
    k5_final           <<<32, 32, 0, stream>>>(hs, ctx, W2, av);
}